// SpectralConv2d_68590627717180
// MI455X (gfx1250) — compile-verified
//
#include <hip/hip_runtime.h>
#include <hip/hip_bf16.h>

typedef __attribute__((ext_vector_type(2))) float v2f;
typedef __attribute__((ext_vector_type(8))) float v8f;

#define B_  8
#define CIN_ 32
#define COUT_ 32
#define M1_ 32
#define M2_ 32
#define MM_ 1024      // M1*M2
#define P_  4096
#define EMB_ 128
#define TWO_PI 6.283185307179586f

// ---------------------------------------------------------------------------
// Kernel 1: phase tables  ph1[b,p,i] = exp(-2pi*i * u * k1_i), k1_i = i-16
// stored interleaved (re,im) float2
// ---------------------------------------------------------------------------
__global__ void phase_kernel(const float* __restrict__ pos,
                             float* __restrict__ ph1, float* __restrict__ ph2) {
    int idx = blockIdx.x * blockDim.x + threadIdx.x;   // b*P + p
    if (idx >= B_ * P_) return;
    float u = pos[(size_t)idx * 2 + 0];
    float v = pos[(size_t)idx * 2 + 1];
    size_t base = (size_t)idx * M1_ * 2;
    #pragma unroll 4
    for (int i = 0; i < M1_; ++i) {
        float k = (float)(i - M1_ / 2);
        float s, c;
        __sincosf(-TWO_PI * u * k, &s, &c);
        ph1[base + 2 * i + 0] = c;
        ph1[base + 2 * i + 1] = s;
        __sincosf(-TWO_PI * v * k, &s, &c);
        ph2[base + 2 * i + 0] = c;
        ph2[base + 2 * i + 1] = s;
    }
}

// ---------------------------------------------------------------------------
// Kernel 2: phi[b,m] = sum_e emb_mod[m,e] * emb[b,e]   (complex * real)
// ---------------------------------------------------------------------------
__global__ void phi_kernel(const float* __restrict__ emb,
                           const float* __restrict__ emb_mod,
                           float* __restrict__ phi) {
    int idx = blockIdx.x * blockDim.x + threadIdx.x;   // b*MM + m
    if (idx >= B_ * MM_) return;
    int b = idx >> 10;
    int m = idx & (MM_ - 1);
    float ar = 0.f, ai = 0.f;
    const float* em = emb_mod + (size_t)m * EMB_ * 2;
    const float* eb = emb + (size_t)b * EMB_;
    for (int e = 0; e < EMB_; ++e) {
        float ev = eb[e];
        ar += em[2 * e + 0] * ev;
        ai += em[2 * e + 1] * ev;
    }
    phi[(size_t)idx * 2 + 0] = ar;
    phi[(size_t)idx * 2 + 1] = ai;
}

// ---------------------------------------------------------------------------
// Kernel 3: forward transform (WMMA f32 16x16x4)
//   x_hat[b,c,m] = sum_p x[b,c,p] * conj(ph1[b,p,i]*ph2[b,p,j]),  m=i*32+j
// One wave computes a 16(c) x 16(mode) tile, re+im accumulators.
// grid = (MM/16, CIN/16, B), block = 32
// ---------------------------------------------------------------------------
__global__ __launch_bounds__(32)
void fwd_kernel(const float* __restrict__ x,
                const float* __restrict__ ph1, const float* __restrict__ ph2,
                float* __restrict__ xhat_re, float* __restrict__ xhat_im) {
    int lane = threadIdx.x;
    int half = lane >> 4;          // selects K pair (A & B operands)
    int lo   = lane & 15;          // A row (c) / B col (mode)
    int b = blockIdx.z, cblk = blockIdx.y, nblk = blockIdx.x;

    int mode = nblk * 16 + lo;
    int mi = mode >> 5, mj = mode & 31;

    const float* xrow = x + ((size_t)b * CIN_ + cblk * 16 + lo) * P_;
    const float* P1 = ph1 + (size_t)b * P_ * M1_ * 2;
    const float* P2 = ph2 + (size_t)b * P_ * M2_ * 2;

    v8f accRe = {};
    v8f accIm = {};
    for (int p0 = 0; p0 < P_; p0 += 4) {
        int kb = p0 + 2 * half;
        v2f a;
        a.x = xrow[kb];
        a.y = xrow[kb + 1];
        v2f bre, bim;
        {
            int pp = kb;
            float c1 = P1[((size_t)pp * M1_ + mi) * 2], s1 = P1[((size_t)pp * M1_ + mi) * 2 + 1];
            float c2 = P2[((size_t)pp * M2_ + mj) * 2], s2 = P2[((size_t)pp * M2_ + mj) * 2 + 1];
            bre.x = c1 * c2 - s1 * s2;
            bim.x = -(c1 * s2 + s1 * c2);          // conjugate
        }
        {
            int pp = kb + 1;
            float c1 = P1[((size_t)pp * M1_ + mi) * 2], s1 = P1[((size_t)pp * M1_ + mi) * 2 + 1];
            float c2 = P2[((size_t)pp * M2_ + mj) * 2], s2 = P2[((size_t)pp * M2_ + mj) * 2 + 1];
            bre.y = c1 * c2 - s1 * s2;
            bim.y = -(c1 * s2 + s1 * c2);
        }
        accRe = __builtin_amdgcn_wmma_f32_16x16x4_f32(false, a, false, bre, (short)0, accRe, false, false);
        accIm = __builtin_amdgcn_wmma_f32_16x16x4_f32(false, a, false, bim, (short)0, accIm, false, false);
    }

    // D layout: N = lo, M = 8*half + v
    #pragma unroll
    for (int v = 0; v < 8; ++v) {
        int c = cblk * 16 + 8 * half + v;
        size_t o = ((size_t)b * CIN_ + c) * MM_ + mode;
        xhat_re[o] = accRe[v];
        xhat_im[o] = accIm[v];
    }
}

// ---------------------------------------------------------------------------
// Kernel 4: channel mixing + phi modulation
//   mod[b,o,m] = phi[b,m] * sum_c x_hat[b,c,m] * w[c,o,m]
// ---------------------------------------------------------------------------
__global__ void mix_kernel(const float* __restrict__ xhat_re,
                           const float* __restrict__ xhat_im,
                           const float* __restrict__ w,       // complex interleaved
                           const float* __restrict__ phi,     // complex interleaved
                           float* __restrict__ mod_re, float* __restrict__ mod_im) {
    int idx = blockIdx.x * blockDim.x + threadIdx.x;  // b*32*MM + o*MM + m
    if (idx >= B_ * COUT_ * MM_) return;
    int b = idx >> 15;
    int o = (idx >> 10) & 31;
    int m = idx & (MM_ - 1);
    float ar = 0.f, ai = 0.f;
    for (int c = 0; c < CIN_; ++c) {
        size_t xo = ((size_t)b * CIN_ + c) * MM_ + m;
        float xr = xhat_re[xo];
        float xi = xhat_im[xo];
        const float* wp = w + (((size_t)c * COUT_ + o) * MM_ + m) * 2;
        float wr = wp[0], wi = wp[1];
        ar += xr * wr - xi * wi;
        ai += xr * wi + xi * wr;
    }
    float pr = phi[((size_t)b * MM_ + m) * 2 + 0];
    float pi = phi[((size_t)b * MM_ + m) * 2 + 1];
    mod_re[idx] = ar * pr - ai * pi;
    mod_im[idx] = ar * pi + ai * pr;
}

// ---------------------------------------------------------------------------
// Kernel 5: inverse transform (WMMA f32 16x16x4)
//   out[b,c,p] = (2/P) * Re( sum_m mod[b,c,m] * ph1[b,p,i]*ph2[b,p,j] )
//             = (2/P) * ( mod_re . PHre  -  mod_im . PHim )
// One wave: 16(c) x 16(p) tile; K over modes. grid = (P/16, CIN/16, B)
// ---------------------------------------------------------------------------
__global__ __launch_bounds__(32)
void inv_kernel(const float* __restrict__ mod_re, const float* __restrict__ mod_im,
                const float* __restrict__ ph1, const float* __restrict__ ph2,
                float* __restrict__ out) {
    int lane = threadIdx.x;
    int half = lane >> 4;
    int lo   = lane & 15;
    int b = blockIdx.z, cblk = blockIdx.y, pblk = blockIdx.x;

    int p = pblk * 16 + lo;                    // B operand column
    const float* P1 = ph1 + ((size_t)b * P_ + p) * M1_ * 2;
    const float* P2 = ph2 + ((size_t)b * P_ + p) * M2_ * 2;
    const float* mre = mod_re + ((size_t)b * COUT_ + cblk * 16 + lo) * MM_;
    const float* mim = mod_im + ((size_t)b * COUT_ + cblk * 16 + lo) * MM_;

    v8f acc = {};
    for (int m0 = 0; m0 < MM_; m0 += 4) {
        int kb = m0 + 2 * half;
        v2f are, aim, bre, bim;
        are.x = mre[kb];
        are.y = mre[kb + 1];
        aim.x = -mim[kb];                      // negate: Re = rr - ii
        aim.y = -mim[kb + 1];
        {
            int m = kb;
            int mi = m >> 5, mj = m & 31;
            float c1 = P1[2 * mi], s1 = P1[2 * mi + 1];
            float c2 = P2[2 * mj], s2 = P2[2 * mj + 1];
            bre.x = c1 * c2 - s1 * s2;
            bim.x = c1 * s2 + s1 * c2;
        }
        {
            int m = kb + 1;
            int mi = m >> 5, mj = m & 31;
            float c1 = P1[2 * mi], s1 = P1[2 * mi + 1];
            float c2 = P2[2 * mj], s2 = P2[2 * mj + 1];
            bre.y = c1 * c2 - s1 * s2;
            bim.y = c1 * s2 + s1 * c2;
        }
        acc = __builtin_amdgcn_wmma_f32_16x16x4_f32(false, are, false, bre, (short)0, acc, false, false);
        acc = __builtin_amdgcn_wmma_f32_16x16x4_f32(false, aim, false, bim, (short)0, acc, false, false);
    }

    const float scale = 2.0f / (float)P_;
    #pragma unroll
    for (int v = 0; v < 8; ++v) {
        int c = cblk * 16 + 8 * half + v;
        out[((size_t)b * CIN_ + c) * P_ + p] = acc[v] * scale;
    }
}

// ---------------------------------------------------------------------------
extern "C" void kernel_launch(void* const* d_in, const int* in_sizes, int n_in,
                              void* d_out, int out_size, void* d_ws, size_t ws_size,
                              hipStream_t stream) {
    const float* x       = (const float*)d_in[0];   // (B,CIN,P)
    const float* pos     = (const float*)d_in[1];   // (B,1,P,2)
    const float* emb     = (const float*)d_in[2];   // (B,EMB)
    const float* weights = (const float*)d_in[3];   // (CIN,COUT,M1,M2) complex64
    const float* emb_mod = (const float*)d_in[4];   // (1,M1,M2,EMB) complex64
    float* out = (float*)d_out;

    // workspace layout (floats)
    float* ws = (float*)d_ws;
    const size_t PH_SZ   = (size_t)B_ * P_ * M1_ * 2;   // 2,097,152
    const size_t PHI_SZ  = (size_t)B_ * MM_ * 2;        // 16,384
    const size_t PLANE   = (size_t)B_ * CIN_ * MM_;     // 262,144
    float* ph1     = ws;
    float* ph2     = ph1 + PH_SZ;
    float* phi     = ph2 + PH_SZ;
    float* xhat_re = phi + PHI_SZ;
    float* xhat_im = xhat_re + PLANE;
    float* mod_re  = xhat_im + PLANE;
    float* mod_im  = mod_re + PLANE;

    // 1) phase tables
    {
        int n = B_ * P_;
        phase_kernel<<<(n + 255) / 256, 256, 0, stream>>>(pos, ph1, ph2);
    }
    // 2) phi
    {
        int n = B_ * MM_;
        phi_kernel<<<(n + 255) / 256, 256, 0, stream>>>(emb, emb_mod, phi);
    }
    // 3) forward WMMA transform
    {
        dim3 grid(MM_ / 16, CIN_ / 16, B_);
        fwd_kernel<<<grid, 32, 0, stream>>>(x, ph1, ph2, xhat_re, xhat_im);
    }
    // 4) mixing + modulation
    {
        int n = B_ * COUT_ * MM_;
        mix_kernel<<<(n + 255) / 256, 256, 0, stream>>>(xhat_re, xhat_im, weights, phi,
                                                        mod_re, mod_im);
    }
    // 5) inverse WMMA transform
    {
        dim3 grid(P_ / 16, CIN_ / 16, B_);
        inv_kernel<<<grid, 32, 0, stream>>>(mod_re, mod_im, ph1, ph2, out);
    }
}